// MCGRU_38551626449013
// MI455X (gfx1250) — compile-verified
//
#include <hip/hip_runtime.h>
#include <math.h>

typedef __attribute__((ext_vector_type(2))) float v2f;
typedef __attribute__((ext_vector_type(8))) float v8f;

#define BS   1024
#define TT   128
#define LAB  64
#define DEMO 16
#define HID  32
#define FDIM 4
#define CATW (HID + LAB * FDIM)   // 288

// ---------------------------------------------------------------------------
// Generic WMMA fp32 GEMM tile kernel:  C[M,N] = A[M,K] @ W[N,K]^T + bias[N]
// One wave32 computes one 16x16 output tile with V_WMMA_F32_16X16X4_F32,
// iterating K in steps of 4. M,N multiples of 16; K multiple of 4.
//
// A-operand layout (16x4 f32): lane&15 selects row M; lanes>=16 hold K+2,K+3
// in {v0,v1}; lanes<16 hold K+0,K+1.  B-operand mirrors with N = lane&15.
// C/D layout: VGPR r, lane L -> row = 8*(L>>4) + r, col = L&15.
// ---------------------------------------------------------------------------
__global__ __launch_bounds__(256) void gemm_nt_bias_wmma(
    const float* __restrict__ A, const float* __restrict__ W,
    const float* __restrict__ bias, float* __restrict__ C,
    int M, int N, int K, int ldc)
{
    const int lane  = threadIdx.x & 31;
    const int wave  = blockIdx.x * (blockDim.x >> 5) + (threadIdx.x >> 5);
    const int ntn   = N >> 4;
    const int ntile = (M >> 4) * ntn;
    if (wave >= ntile) return;            // wave-uniform: EXEC stays all-ones

    const int tm   = wave / ntn;
    const int tn   = wave % ntn;
    const int half = lane >> 4;           // 0: K+{0,1}, 1: K+{2,3}
    const int l15  = lane & 15;

    const float* Ap = A + (size_t)(tm * 16 + l15) * K + 2 * half;
    const float* Wp = W + (size_t)(tn * 16 + l15) * K + 2 * half;

    v8f c = {};
    for (int k = 0; k < K; k += 4) {
        v2f a = *(const v2f*)(Ap + k);
        v2f b = *(const v2f*)(Wp + k);
        c = __builtin_amdgcn_wmma_f32_16x16x4_f32(
                /*neg_a=*/false, a, /*neg_b=*/false, b,
                /*c_mod=*/(short)0, c, /*reuse_a=*/false, /*reuse_b=*/false);
    }

    const float bv = bias[tn * 16 + l15];                 // per-column bias
    float* Cp = C + (size_t)(tm * 16 + 8 * half) * ldc + (tn * 16 + l15);
#pragma unroll
    for (int r = 0; r < 8; ++r)
        Cp[(size_t)r * ldc] = c[r] + bv;
}

// ---------------------------------------------------------------------------
// GRU scan: one thread per (batch, lab) pair; hidden F=4 lives in registers,
// all per-lab weights (12 + 48 + 16 floats) register-resident.
// xp layout: [BS*T, LAB] so lane-adjacent threads (same b, adjacent l) load
// coalesced 32-bit words each step. Writes final h into cat[:, 32 + l*4 + j].
// ---------------------------------------------------------------------------
__device__ __forceinline__ float sigmoidf(float x) {
    return 1.0f / (1.0f + __expf(-x));
}

__global__ __launch_bounds__(256) void mcgru_scan(
    const float* __restrict__ xp,   // [BS*T, LAB]
    const float* __restrict__ Wih,  // [LAB, 12]
    const float* __restrict__ bih,  // [LAB, 12]
    const float* __restrict__ Whh,  // [LAB, 12, 4]
    const float* __restrict__ bhh,  // [LAB, 12]
    float* __restrict__ cat)        // [BS, 288]
{
    const int tid = blockIdx.x * blockDim.x + threadIdx.x;  // b*LAB + l
    const int l   = tid & (LAB - 1);
    const int b   = tid >> 6;
    if (b >= BS) return;

    float wih[12], whh[12][4], brz[8], bin[4], bhn[4];
#pragma unroll
    for (int g = 0; g < 12; ++g) {
        wih[g] = Wih[l * 12 + g];
#pragma unroll
        for (int j = 0; j < 4; ++j) whh[g][j] = Whh[l * 48 + g * 4 + j];
    }
#pragma unroll
    for (int g = 0; g < 8; ++g) brz[g] = bih[l * 12 + g] + bhh[l * 12 + g];
#pragma unroll
    for (int j = 0; j < 4; ++j) {
        bin[j] = bih[l * 12 + 8 + j];
        bhn[j] = bhh[l * 12 + 8 + j];
    }

    float h[4] = {0.f, 0.f, 0.f, 0.f};
    const float* xrow = xp + (size_t)b * TT * LAB + l;

    for (int t = 0; t < TT; ++t) {
        const float xv = xrow[(size_t)t * LAB];
        if (t + 1 < TT)
            __builtin_prefetch(xrow + (size_t)(t + 1) * LAB, 0, 0);

        float gh[12];
#pragma unroll
        for (int g = 0; g < 12; ++g)
            gh[g] = h[0] * whh[g][0] + h[1] * whh[g][1]
                  + h[2] * whh[g][2] + h[3] * whh[g][3];

#pragma unroll
        for (int j = 0; j < 4; ++j) {
            const float rg = sigmoidf(xv * wih[j]     + gh[j]     + brz[j]);
            const float zg = sigmoidf(xv * wih[4 + j] + gh[4 + j] + brz[4 + j]);
            const float ng = tanhf(xv * wih[8 + j] + bin[j] + rg * (gh[8 + j] + bhn[j]));
            h[j] = (1.0f - zg) * ng + zg * h[j];
        }
    }

    float* dst = cat + (size_t)b * CATW + HID + l * 4;
#pragma unroll
    for (int j = 0; j < 4; ++j) dst[j] = h[j];
}

// ---------------------------------------------------------------------------
extern "C" void kernel_launch(void* const* d_in, const int* in_sizes, int n_in,
                              void* d_out, int out_size, void* d_ws, size_t ws_size,
                              hipStream_t stream) {
    const float* x      = (const float*)d_in[0];
    const float* statc  = (const float*)d_in[1];
    const float* demo_W = (const float*)d_in[2];
    const float* demo_b = (const float*)d_in[3];
    const float* lab_W  = (const float*)d_in[4];
    const float* lab_b  = (const float*)d_in[5];
    const float* Wih    = (const float*)d_in[6];
    const float* bih    = (const float*)d_in[7];
    const float* Whh    = (const float*)d_in[8];
    const float* bhh    = (const float*)d_in[9];
    const float* out_W  = (const float*)d_in[10];
    const float* out_b  = (const float*)d_in[11];
    float*       out    = (float*)d_out;

    float* xp  = (float*)d_ws;                                   // [BS*T, LAB]
    float* cat = (float*)((char*)d_ws +
                          (size_t)BS * TT * LAB * sizeof(float)); // [BS, 288]

    // 1) xp = x @ lab_W^T + lab_b   (M=131072, N=64, K=64)
    {
        const int tiles  = (BS * TT / 16) * (LAB / 16);  // 32768
        const int blocks = (tiles + 7) / 8;
        gemm_nt_bias_wmma<<<blocks, 256, 0, stream>>>(
            x, lab_W, lab_b, xp, BS * TT, LAB, LAB, LAB);
    }

    // 2) cat[:, :32] = static @ demo_W^T + demo_b   (M=1024, N=32, K=16)
    {
        const int tiles  = (BS / 16) * (HID / 16);       // 128
        const int blocks = (tiles + 7) / 8;
        gemm_nt_bias_wmma<<<blocks, 256, 0, stream>>>(
            statc, demo_W, demo_b, cat, BS, HID, DEMO, CATW);
    }

    // 3) GRU scan -> cat[:, 32:288]
    mcgru_scan<<<(BS * LAB) / 256, 256, 0, stream>>>(
        xp, Wih, bih, Whh, bhh, cat);

    // 4) out = cat @ out_W^T + out_b   (M=1024, N=32, K=288)
    {
        const int tiles  = (BS / 16) * (HID / 16);       // 128
        const int blocks = (tiles + 7) / 8;
        gemm_nt_bias_wmma<<<blocks, 256, 0, stream>>>(
            cat, out_W, out_b, out, BS, HID, CATW, HID);
    }
}